// PSHGCN_32126355374617
// MI455X (gfx1250) — compile-verified
//
#include <hip/hip_runtime.h>
#include <hip/hip_bf16.h>

// Problem constants (match the reference)
#define N_A   100000
#define N_P   150000
#define NTOT  250000          // N_A + N_P, multiple of 16
#define DDIM  128
#define CDIM  16
#define EDG   2000000

typedef float v2f __attribute__((ext_vector_type(2)));
typedef float v8f __attribute__((ext_vector_type(8)));
typedef int   v4i __attribute__((vector_size(16)));   // matches builtin param type

#if __has_builtin(__builtin_amdgcn_global_load_async_to_lds_b128)
#define HAVE_ASYNC_LDS 1
#else
#define HAVE_ASYNC_LDS 0
#endif

// ---- WMMA f32 16x16x4 fragment helpers (per CDNA5 ISA VGPR layouts) ----
// A (16x4, row-major, stride lda): lanes 0-15 -> K=0,1 ; lanes 16-31 -> K=2,3
__device__ __forceinline__ v2f load_a_frag(const float* A, int lda, int k, int r, int half) {
    const float* p = A + r * lda + k + 2 * half;
    v2f a; a.x = p[0]; a.y = p[1];
    return a;
}
// B (4x16, row-major, stride ldb, caller folds column base into B):
// VGPR0: lanes0-15 K=0, lanes16-31 K=2 ; VGPR1: K=1 / K=3
__device__ __forceinline__ v2f load_b_frag(const float* B, int ldb, int k, int r, int half) {
    v2f b;
    b.x = B[(k + 2 * half)     * ldb + r];
    b.y = B[(k + 2 * half + 1) * ldb + r];
    return b;
}

// ============================================================================
// Kernel 1: fused  X = normalize(relu(concat(x0@Wp0, x1@Wp1) @ W1 + b1))
// One block = 16 output rows; 8 waves each own a 16-col tile; strips in LDS.
// ============================================================================
__global__ void __launch_bounds__(256) proj_norm_kernel(
    const float* __restrict__ x0, const float* __restrict__ x1,
    const float* __restrict__ Wp0, const float* __restrict__ Wp1,
    const float* __restrict__ W1,  const float* __restrict__ b1,
    float* __restrict__ outX)
{
    __shared__ float sA[16 * DDIM];
    __shared__ float sY[16 * DDIM];
    __shared__ float sMean[16];
    __shared__ float sRstd[16];

    const int tid  = threadIdx.x;
    const int wave = tid >> 5;
    const int lane = tid & 31;
    const int r    = lane & 15;
    const int half = lane >> 4;
    const int m0   = blockIdx.x * 16;       // global row tile base

    const float* xsel; const float* Wsel; int row0;
    if (m0 < N_A) { xsel = x0; Wsel = Wp0; row0 = m0; }
    else          { xsel = x1; Wsel = Wp1; row0 = m0 - N_A; }

    // Stage the 16x128 input strip (contiguous 8 KB) into LDS
    const float* src = xsel + (size_t)row0 * DDIM;
#if HAVE_ASYNC_LDS
    // gfx1250 async global->LDS DMA (ASYNCcnt-tracked, bypasses VGPRs)
    for (int i = tid; i < (16 * DDIM) / 4; i += 256) {
        __builtin_amdgcn_global_load_async_to_lds_b128(
            (__attribute__((address_space(1))) v4i*)(src + (size_t)i * 4),
            (__attribute__((address_space(3))) v4i*)(sA + i * 4),
            0, 0);
    }
#if __has_builtin(__builtin_amdgcn_s_wait_asynccnt)
    __builtin_amdgcn_s_wait_asynccnt(0);
#else
    asm volatile("s_wait_asynccnt 0" ::: "memory");
#endif
#else
    for (int i = tid; i < 16 * DDIM; i += 256) sA[i] = src[i];
#endif
    __syncthreads();

    // GEMM1: sY = sA @ Wsel  (wave owns cols [16w, 16w+16))
    {
        const int n0 = wave * 16;
        v8f acc = {0.f,0.f,0.f,0.f,0.f,0.f,0.f,0.f};
        for (int k = 0; k < DDIM; k += 4) {
            v2f a = load_a_frag(sA, DDIM, k, r, half);
            v2f b = load_b_frag(Wsel + n0, DDIM, k, r, half);
            acc = __builtin_amdgcn_wmma_f32_16x16x4_f32(false, a, false, b,
                                                        (short)0, acc, false, false);
        }
#pragma unroll
        for (int i = 0; i < 8; ++i)
            sY[(i + 8 * half) * DDIM + n0 + r] = acc[i];
    }
    __syncthreads();

    // GEMM2: Z = relu(sY @ W1 + b1) -> reuse sA
    {
        const int n0 = wave * 16;
        v8f acc = {0.f,0.f,0.f,0.f,0.f,0.f,0.f,0.f};
        for (int k = 0; k < DDIM; k += 4) {
            v2f a = load_a_frag(sY, DDIM, k, r, half);
            v2f b = load_b_frag(W1 + n0, DDIM, k, r, half);
            acc = __builtin_amdgcn_wmma_f32_16x16x4_f32(false, a, false, b,
                                                        (short)0, acc, false, false);
        }
        const float bias = b1[n0 + r];
#pragma unroll
        for (int i = 0; i < 8; ++i) {
            float z = acc[i] + bias;
            sA[(i + 8 * half) * DDIM + n0 + r] = z > 0.f ? z : 0.f;
        }
    }
    __syncthreads();

    // Row statistics (mean, unbiased std with ddof=1)
    if (tid < 16) {
        float s = 0.f;
        for (int c = 0; c < DDIM; ++c) s += sA[tid * DDIM + c];
        float mean = s * (1.0f / DDIM);
        float ss = 0.f;
        for (int c = 0; c < DDIM; ++c) { float d = sA[tid * DDIM + c] - mean; ss += d * d; }
        sMean[tid] = mean;
        sRstd[tid] = 1.0f / sqrtf(ss * (1.0f / (DDIM - 1)));
    }
    __syncthreads();

    for (int i = tid; i < 16 * DDIM; i += 256) {
        int rr = i >> 7;
        float z = (sA[i] - sMean[rr]) * sRstd[rr];
        if (__builtin_isnan(z)) z = 0.f;
        outX[(size_t)m0 * DDIM + i] = z;
    }
}

// ============================================================================
// Kernel 2: zero the two hop-1 accumulators (float4 streaming)
// ============================================================================
__global__ void __launch_bounds__(256) zero2_kernel(
    float4* __restrict__ h1, float4* __restrict__ h2)
{
    size_t i = (size_t)blockIdx.x * 256 + threadIdx.x;
    const float4 z = {0.f, 0.f, 0.f, 0.f};
    h1[i] = z;
    h2[i] = z;
}

// ============================================================================
// Kernel 3: streaming combine (no atomics):
//   res = coe0*src + coe1*h1 + coe2*h2
// ============================================================================
__global__ void __launch_bounds__(256) combine_kernel(
    const float4* __restrict__ src, const float4* __restrict__ h1,
    const float4* __restrict__ h2, float4* __restrict__ res,
    const float* __restrict__ coe)
{
    size_t i = (size_t)blockIdx.x * 256 + threadIdx.x;
    const float c0 = coe[0], c1 = coe[1], c2 = coe[2];
    float4 s = src[i], a = h1[i], b = h2[i], o;
    o.x = c0 * s.x + c1 * a.x + c2 * b.x;
    o.y = c0 * s.y + c1 * a.y + c2 * b.y;
    o.z = c0 * s.z + c1 * a.z + c2 * b.z;
    o.w = c0 * s.w + c1 * a.w + c2 * b.w;
    res[i] = o;
}

// ============================================================================
// Kernel 4: SpMM, one wave per edge, lane handles 4 contiguous features:
//   dst[row] += scale * v * src[col]     (scale = coe[cidx] if cidx>=0 else 1)
// ============================================================================
__global__ void __launch_bounds__(256) spmm_kernel(
    const int* __restrict__ rows, const int* __restrict__ cols,
    const float* __restrict__ vals, const float* __restrict__ src,
    float* __restrict__ dst, const float* __restrict__ coe, int cidx, int nE)
{
    int gw   = (int)(((size_t)blockIdx.x * blockDim.x + threadIdx.x) >> 5);
    int lane = threadIdx.x & 31;
    if (gw >= nE) return;

    int   row = rows[gw];
    int   col = cols[gw];
    float v   = vals[gw];
    if (cidx >= 0) v *= coe[cidx];

    const float4 xv = *(const float4*)(src + (size_t)col * DDIM + lane * 4);
    size_t ob = (size_t)row * DDIM + lane * 4;

    unsafeAtomicAdd(dst + ob + 0, v * xv.x);
    unsafeAtomicAdd(dst + ob + 1, v * xv.y);
    unsafeAtomicAdd(dst + ob + 2, v * xv.z);
    unsafeAtomicAdd(dst + ob + 3, v * xv.w);
}

// ============================================================================
// Kernel 5: out = X @ W2 + b2   ([N,128] @ [128,16]); one wave per 16-row tile
// ============================================================================
__global__ void __launch_bounds__(256) out_gemm_kernel(
    const float* __restrict__ X, const float* __restrict__ W2,
    const float* __restrict__ b2, float* __restrict__ out, int ntiles)
{
    const int wave = threadIdx.x >> 5;
    const int lane = threadIdx.x & 31;
    const int r    = lane & 15;
    const int half = lane >> 4;
    const int tile = blockIdx.x * 8 + wave;
    if (tile >= ntiles) return;                 // wave-uniform

    const float* A = X + (size_t)tile * 16 * DDIM;
    v8f acc = {0.f,0.f,0.f,0.f,0.f,0.f,0.f,0.f};
    for (int k = 0; k < DDIM; k += 4) {
        v2f a = load_a_frag(A, DDIM, k, r, half);
        v2f b = load_b_frag(W2, CDIM, k, r, half);
        acc = __builtin_amdgcn_wmma_f32_16x16x4_f32(false, a, false, b,
                                                    (short)0, acc, false, false);
    }
    const float bias = b2[r];
#pragma unroll
    for (int i = 0; i < 8; ++i)
        out[(size_t)(tile * 16 + i + 8 * half) * CDIM + r] = acc[i] + bias;
}

// ============================================================================
extern "C" void kernel_launch(void* const* d_in, const int* in_sizes, int n_in,
                              void* d_out, int out_size, void* d_ws, size_t ws_size,
                              hipStream_t stream) {
    const float* x0      = (const float*)d_in[0];
    const float* x1      = (const float*)d_in[1];
    const float* vals_ap = (const float*)d_in[2];
    const float* vals_pa = (const float*)d_in[3];
    const int*   rows_ap = (const int*)d_in[4];
    const int*   cols_ap = (const int*)d_in[5];
    const int*   rows_pa = (const int*)d_in[6];
    const int*   cols_pa = (const int*)d_in[7];
    const float* Wp0     = (const float*)d_in[8];
    const float* Wp1     = (const float*)d_in[9];
    const float* W1      = (const float*)d_in[10];
    const float* b1      = (const float*)d_in[11];
    const float* W2      = (const float*)d_in[12];
    const float* b2      = (const float*)d_in[13];
    const float* coe     = (const float*)d_in[14];
    float* out = (float*)d_out;

    const size_t bufElems = (size_t)NTOT * DDIM;     // 32M floats = 128 MB
    float* bufX  = (float*)d_ws;
    float* bufH1 = bufX  + bufElems;
    float* bufH2 = bufH1 + bufElems;
    float* bufR  = bufH2 + bufElems;

    const int projBlocks = NTOT / 16;                // 15625
    const int vec4Blocks = (int)(bufElems / 4 / 256);// 31250  (float4 streaming)
    const int spmmBlocks = EDG / 8;                  // 250000 (8 waves/block, 1 edge/wave)
    const int outTiles   = NTOT / 16;

    // Feature extraction: X = normalize(relu(concat(x0@Wp0, x1@Wp1)@W1 + b1))
    proj_norm_kernel<<<projBlocks, 256, 0, stream>>>(x0, x1, Wp0, Wp1, W1, b1, bufX);

    // ---- power pass 1: first = AP (h1), second = PA (h2); src = X, res = R
    zero2_kernel<<<vec4Blocks, 256, 0, stream>>>((float4*)bufH1, (float4*)bufH2);
    spmm_kernel<<<spmmBlocks, 256, 0, stream>>>(rows_ap, cols_ap, vals_ap, bufX, bufH1, coe, -1, EDG);
    spmm_kernel<<<spmmBlocks, 256, 0, stream>>>(rows_pa, cols_pa, vals_pa, bufX, bufH2, coe, -1, EDG);
    combine_kernel<<<vec4Blocks, 256, 0, stream>>>((const float4*)bufX, (const float4*)bufH1,
                                                   (const float4*)bufH2, (float4*)bufR, coe);
    spmm_kernel<<<spmmBlocks, 256, 0, stream>>>(rows_ap, cols_ap, vals_ap, bufH2, bufR, coe, 3, EDG);
    spmm_kernel<<<spmmBlocks, 256, 0, stream>>>(rows_pa, cols_pa, vals_pa, bufH1, bufR, coe, 4, EDG);

    // ---- power pass 2: first = PA (h1), second = AP (h2); src = R, res = X
    zero2_kernel<<<vec4Blocks, 256, 0, stream>>>((float4*)bufH1, (float4*)bufH2);
    spmm_kernel<<<spmmBlocks, 256, 0, stream>>>(rows_pa, cols_pa, vals_pa, bufR, bufH1, coe, -1, EDG);
    spmm_kernel<<<spmmBlocks, 256, 0, stream>>>(rows_ap, cols_ap, vals_ap, bufR, bufH2, coe, -1, EDG);
    combine_kernel<<<vec4Blocks, 256, 0, stream>>>((const float4*)bufR, (const float4*)bufH1,
                                                   (const float4*)bufH2, (float4*)bufX, coe);
    spmm_kernel<<<spmmBlocks, 256, 0, stream>>>(rows_pa, cols_pa, vals_pa, bufH2, bufX, coe, 3, EDG);
    spmm_kernel<<<spmmBlocks, 256, 0, stream>>>(rows_ap, cols_ap, vals_ap, bufH1, bufX, coe, 4, EDG);

    // Classifier: out = resX @ W2 + b2
    out_gemm_kernel<<<(outTiles + 7) / 8, 256, 0, stream>>>(bufX, W2, b2, out, outTiles);
}